// TwoLayerGraphSage_20710332301832
// MI455X (gfx1250) — compile-verified
//
#include <hip/hip_runtime.h>
#include <hip/hip_bf16.h>

// ---------------------------------------------------------------------------
// Two-layer GraphSAGE (mean) for MI455X / gfx1250 (wave32, WMMA).
//   layer: out = x @ W_self + (scatter_mean(x[src] -> dst)) @ W_neigh + b
//   layer1 followed by ELU, layer2 raw.
// GEMMs use V_WMMA_F32_16X16X4_F32 (full fp32 matrix path), with TWO
// independent accumulator chains (self / neigh) to double WMMA ILP.
// Scatter uses native global_atomic_add_f32 (L2-resident: working set < 192MB L2).
// ---------------------------------------------------------------------------

typedef __attribute__((ext_vector_type(2))) float v2f;
typedef __attribute__((ext_vector_type(8))) float v8f;

__device__ __forceinline__ v8f wmma_f32_16x16x4(v2f a, v2f b, v8f c) {
  // 8 args: (neg_a, A, neg_b, B, c_mod, C, reuse_a, reuse_b)
  return __builtin_amdgcn_wmma_f32_16x16x4_f32(false, a, false, b, (short)0, c,
                                               false, false);
}

// ---------------------------------------------------------------------------
__global__ void zero_f32_kernel(float* __restrict__ p, long n) {
  long i = (long)blockIdx.x * blockDim.x + threadIdx.x;
  long stride = (long)gridDim.x * blockDim.x;
  for (; i < n; i += stride) p[i] = 0.0f;
}

// ---------------------------------------------------------------------------
// One edge per wave32: each lane handles 4 consecutive floats (float4 load,
// 4 fp32 atomic adds into the destination row). Lane 0 optionally bumps deg.
__global__ void scatter_kernel(const float* __restrict__ feat,
                               const int* __restrict__ src,
                               const int* __restrict__ dst,
                               float* __restrict__ msg,
                               float* __restrict__ deg, int E) {
  int e = blockIdx.x * 8 + (threadIdx.x >> 5);
  if (e >= E) return;
  int lane = threadIdx.x & 31;
  int s = src[e];
  int d = dst[e];
  const float4 v = *(const float4*)(feat + (size_t)s * 128 + lane * 4);
  float* p = msg + (size_t)d * 128 + lane * 4;
  atomicAdd(p + 0, v.x);
  atomicAdd(p + 1, v.y);
  atomicAdd(p + 2, v.z);
  atomicAdd(p + 3, v.w);
  if (deg != nullptr && lane == 0) atomicAdd(deg + d, 1.0f);
}

// ---------------------------------------------------------------------------
// One block (256 threads = 8 wave32) per 16-row node tile.
// LDS row stride 132 floats -> bank-conflict-free ds_load_b64 A-fragment reads.
// Wave w computes the 16x16 output tile at columns [w*16, w*16+16).
#define LDSS 132

__global__ void sage_gemm_kernel(const float* __restrict__ feat,
                                 const float* __restrict__ msg,
                                 const float* __restrict__ deg,
                                 const float* __restrict__ Wself,
                                 const float* __restrict__ Wneigh,
                                 const float* __restrict__ bias,
                                 float* __restrict__ out,
                                 int N, int apply_elu) {
  __shared__ __align__(16) float xs[16 * LDSS];  // self features tile
  __shared__ __align__(16) float hs[16 * LDSS];  // mean-aggregated tile

  int row0 = blockIdx.x * 16;
  if (row0 >= N) return;  // uniform per block -> EXEC stays all-ones
  int tid = threadIdx.x;

  // Stage 16x128 of x and msg/deg into LDS (2048 elements, 8 per thread).
#pragma unroll
  for (int i = 0; i < 8; ++i) {
    int idx = tid + i * 256;
    int r = idx >> 7;
    int c = idx & 127;
    float dg = deg[row0 + r];
    float inv = (dg > 1.0f) ? (1.0f / dg) : 1.0f;  // max(deg,1)
    xs[r * LDSS + c] = feat[(size_t)(row0 + r) * 128 + c];
    hs[r * LDSS + c] = msg[(size_t)(row0 + r) * 128 + c] * inv;
  }
  __syncthreads();

  int wave = tid >> 5;        // 0..7 : output column tile
  int lane = tid & 31;
  int m = lane & 15;          // A: row M ; B/C/D: col N (same expression)
  int kb = (lane >> 4) << 1;  // K sub-pair selected by half-wave: 0 or 2
  int ncol = wave * 16 + m;   // global output column

  // Two independent accumulator chains: self (seeded with bias) and neigh.
  float bv = bias[ncol];
  v8f accS, accN;
#pragma unroll
  for (int v = 0; v < 8; ++v) { accS[v] = bv; accN[v] = 0.0f; }

  // K loop: 128 / 4 = 32 steps; the two WMMAs per step are independent.
#pragma unroll
  for (int k = 0; k < 128; k += 4) {
    v2f aS = *(const v2f*)(&xs[m * LDSS + k + kb]);  // ds_load_b64
    v2f aN = *(const v2f*)(&hs[m * LDSS + k + kb]);
    v2f bS, bN;
    bS.x = Wself[(size_t)(k + kb) * 128 + ncol];
    bS.y = Wself[(size_t)(k + kb + 1) * 128 + ncol];
    bN.x = Wneigh[(size_t)(k + kb) * 128 + ncol];
    bN.y = Wneigh[(size_t)(k + kb + 1) * 128 + ncol];
    accS = wmma_f32_16x16x4(aS, bS, accS);
    accN = wmma_f32_16x16x4(aN, bN, accN);
  }

  // D layout: VGPR v holds row (v + 8*(lane>=16)), col = lane&15.
  int mrow = (lane >> 4) << 3;
#pragma unroll
  for (int v = 0; v < 8; ++v) {
    float val = accS[v] + accN[v];
    if (apply_elu) val = (val > 0.0f) ? val : (__expf(val) - 1.0f);
    out[(size_t)(row0 + v + mrow) * 128 + ncol] = val;
  }
}

// ---------------------------------------------------------------------------
extern "C" void kernel_launch(void* const* d_in, const int* in_sizes, int n_in,
                              void* d_out, int out_size, void* d_ws, size_t ws_size,
                              hipStream_t stream) {
  const float* x   = (const float*)d_in[0];
  const int* src   = (const int*)d_in[1];
  const int* dst   = (const int*)d_in[2];
  const float* W1s = (const float*)d_in[3];
  const float* W1n = (const float*)d_in[4];
  const float* b1  = (const float*)d_in[5];
  const float* W2s = (const float*)d_in[6];
  const float* W2n = (const float*)d_in[7];
  const float* b2  = (const float*)d_in[8];
  float* out = (float*)d_out;

  const int N = in_sizes[0] / 128;
  const int E = in_sizes[1];

  // Workspace: msg [N*128] | h [N*128] | deg [N]   (~51.4 MB for N=50000)
  float* msg = (float*)d_ws;
  float* h   = msg + (size_t)N * 128;
  float* deg = h + (size_t)N * 128;

  const long nmsg = (long)N * 128;
  dim3 blk(256);
  dim3 zgrid((unsigned)((nmsg + 255) / 256 < 16384 ? (nmsg + 255) / 256 : 16384));
  dim3 sgrid((unsigned)((E + 7) / 8));
  dim3 ggrid((unsigned)((N + 15) / 16));

  // ---- Layer 1 ----
  zero_f32_kernel<<<zgrid, blk, 0, stream>>>(msg, nmsg);
  zero_f32_kernel<<<dim3((N + 255) / 256), blk, 0, stream>>>(deg, (long)N);
  scatter_kernel<<<sgrid, blk, 0, stream>>>(x, src, dst, msg, deg, E);
  sage_gemm_kernel<<<ggrid, blk, 0, stream>>>(x, msg, deg, W1s, W1n, b1, h, N,
                                              /*apply_elu=*/1);

  // ---- Layer 2 ----
  zero_f32_kernel<<<zgrid, blk, 0, stream>>>(msg, nmsg);
  scatter_kernel<<<sgrid, blk, 0, stream>>>(h, src, dst, msg, nullptr, E);
  sage_gemm_kernel<<<ggrid, blk, 0, stream>>>(h, msg, deg, W2s, W2n, b2, out, N,
                                              /*apply_elu=*/0);
}